// OrdinalLoss_49847390437462
// MI455X (gfx1250) — compile-verified
//
#include <hip/hip_runtime.h>
#include <hip/hip_bf16.h>

// ---------------- problem constants (match reference) ----------------
#define NUM_CLASSES 100
#define COLS 101                    // NUM_CLASSES + 1
#define BATCH_N 524288

// ---------------- launch geometry ----------------
#define BLOCKS 2048
#define THREADS 256
#define WAVES_PER_BLOCK (THREADS / 32)
#define TOTAL_WAVES (BLOCKS * WAVES_PER_BLOCK)          // 16384
#define ROWS_PER_WAVE (BATCH_N / TOTAL_WAVES)           // 32
static_assert(ROWS_PER_WAVE == 32, "age-broadcast trick requires 32 rows/wave");
static_assert(BATCH_N == TOTAL_WAVES * ROWS_PER_WAVE, "exact tiling");

#define LOG2E_F 1.4426950408889634f
#define LN2_F   0.6931471805599453f

typedef __attribute__((ext_vector_type(2))) float v2f;
typedef __attribute__((ext_vector_type(8))) float v8f;

// Reduce one float per lane (32 values) to a single wave-wide sum using
// V_WMMA_F32_16X16X4_F32 with B == ones.  Each lane's value lands in exactly
// one A[m][k] slot, so D[m][n] = p_m + p_{m+16} (n-independent).
// C/D layout: lane<16 VGPR j = D[j][n], lane>=16 = D[j+8][n].
__device__ __forceinline__ float wave_sum_wmma(float v, int lane) {
    v2f a; a.x = v; a.y = 0.0f;        // lane -> one A element, K-partner zero
    v2f b; b.x = 1.0f; b.y = 1.0f;     // B = ones(4x16)
    v8f c = {};
    c = __builtin_amdgcn_wmma_f32_16x16x4_f32(
        /*neg_a=*/false, a, /*neg_b=*/false, b,
        /*c_mod=*/(short)0, c, /*reuse_a=*/false, /*reuse_b=*/false);
    float s = c[0] + c[1] + c[2] + c[3] + c[4] + c[5] + c[6] + c[7];
    // lanes 0..15 hold sum of D[0..7], lanes 16..31 hold sum of D[8..15]
    s += __shfl(s, lane ^ 16, 32);     // full wave sum on every lane
    return s;
}

__global__ __launch_bounds__(THREADS)
void ordinal_partials_kernel(const float* __restrict__ logits,
                             const float* __restrict__ ages,
                             float* __restrict__ partials) {
    const int tid  = threadIdx.x;
    const int lane = tid & 31;
    const int wid  = tid >> 5;
    const int waveGlobal = blockIdx.x * WAVES_PER_BLOCK + wid;
    const long long row0 = (long long)waveGlobal * ROWS_PER_WAVE;

    // one coalesced load covers this wave's 32 ages
    const float myAge = ages[row0 + lane];

    float rest_acc = 0.0f;  // per-lane partial of sum(max(x,0) - x*t)
    float log_acc  = 0.0f;  // per-lane partial of sum(log2(1 + 2^(-|x|*log2e)))
    float mae_acc  = 0.0f;  // lane i accumulates row i's |preds - age|

    const float* rp = logits + row0 * COLS;

    for (int i = 0; i < ROWS_PER_WAVE; ++i) {
        // wave-uniform scalar broadcast of this row's age (v_readlane)
        const float age = __uint_as_float(
            __builtin_amdgcn_readlane(__float_as_uint(myAge), i));
        const float ai   = floorf(age);
        const float frac = age - ai;
        const int   iai  = (int)ai;
        const bool  soft = (iai < NUM_CLASSES);

        int pcnt = 0;
#pragma unroll
        for (int j = 0; j < 4; ++j) {
            const int col  = lane + 32 * j;
            const int colc = (col < COLS) ? col : (COLS - 1);   // clamp addr
            float x = __builtin_nontemporal_load(rp + colc);
            x = (col < COLS) ? x : -1.0e30f;  // pad: softplus->0, t->0, cnt->0

            // stable softplus = max(x,0) + ln2*log2(1 + 2^(-|x|*log2e));
            // ln2 factor deferred to the end of the kernel.
            const float e = __builtin_amdgcn_exp2f(-fabsf(x) * LOG2E_F);
            log_acc += __builtin_amdgcn_logf(1.0f + e);

            // ordinal target (pure cndmask chain, no branches)
            const int d = col - iai;
            float t = (d < 0) ? 1.0f : 0.0f;
            t = (soft && d == 0) ? 1.0f - frac : t;
            t = (soft && d == 1) ? frac : t;

            rest_acc += fmaxf(x, 0.0f);
            rest_acc = __builtin_fmaf(-x, t, rest_acc);

            // preds: count logits > 0 across the row (uniform result)
            pcnt += __popcll(__ballot(x > 0.0f));
        }
        mae_acc += (lane == i) ? fabsf((float)pcnt - age) : 0.0f;
        rp += COLS;
    }

    const float bce_acc = __builtin_fmaf(LN2_F, log_acc, rest_acc);

    // wave-level reductions via WMMA (EXEC all-ones here; no divergence)
    const float bce_wave = wave_sum_wmma(bce_acc, lane);
    const float mae_wave = wave_sum_wmma(mae_acc, lane);

    __shared__ float sB[WAVES_PER_BLOCK];
    __shared__ float sM[WAVES_PER_BLOCK];
    if (lane == 0) { sB[wid] = bce_wave; sM[wid] = mae_wave; }
    __syncthreads();

    if (tid == 0) {
        float b = 0.0f, m = 0.0f;
#pragma unroll
        for (int w = 0; w < WAVES_PER_BLOCK; ++w) { b += sB[w]; m += sM[w]; }
        partials[2 * blockIdx.x]     = b;
        partials[2 * blockIdx.x + 1] = m;
    }
}

__global__ __launch_bounds__(256)
void ordinal_finalize_kernel(const float* __restrict__ partials,
                             float* __restrict__ out) {
    __shared__ float sB[256];
    __shared__ float sM[256];
    const int tid = threadIdx.x;
    float b = 0.0f, m = 0.0f;
    for (int j = tid; j < BLOCKS; j += 256) {   // fixed order -> deterministic
        b += partials[2 * j];
        m += partials[2 * j + 1];
    }
    sB[tid] = b; sM[tid] = m;
    __syncthreads();
    for (int off = 128; off > 0; off >>= 1) {
        if (tid < off) { sB[tid] += sB[tid + off]; sM[tid] += sM[tid + off]; }
        __syncthreads();
    }
    if (tid == 0) {
        // B*COLS = 101 * 2^19 is exactly representable in f32
        const float bce = sB[0] * (1.0f / ((float)BATCH_N * (float)COLS));
        const float mae = sM[0] * (1.0f / (float)BATCH_N);
        out[0] = 0.7f * bce + 0.3f * mae;
    }
}

extern "C" void kernel_launch(void* const* d_in, const int* in_sizes, int n_in,
                              void* d_out, int out_size, void* d_ws, size_t ws_size,
                              hipStream_t stream) {
    const float* logits = (const float*)d_in[0];   // [BATCH, 101] f32
    const float* ages   = (const float*)d_in[1];   // [BATCH]      f32
    float* partials = (float*)d_ws;                // needs 2*BLOCKS*4 = 16 KB
    float* out = (float*)d_out;

    ordinal_partials_kernel<<<BLOCKS, THREADS, 0, stream>>>(logits, ages, partials);
    ordinal_finalize_kernel<<<1, 256, 0, stream>>>(partials, out);
}